// QuantumLayer_21500606284062
// MI455X (gfx1250) — compile-verified
//
#include <hip/hip_runtime.h>

typedef float v2f __attribute__((ext_vector_type(2)));
typedef float v8f __attribute__((ext_vector_type(8)));

#define NQ      16
#define NSTATE  65536
#define THREADS 1024

// ---------------------------------------------------------------------------
// One fused 16x16 gate (4 RYs kron'd) applied to the 65536-amplitude state in
// LDS as 256 independent 16x16x16 f32 GEMM tiles, each = 4 chained
// V_WMMA_F32_16X16X4_F32. Amplitude index bit (15-q) <-> qubit q; group G
// covers qubits 4G..4G+3 == index bits [15-4G : 12-4G].
//   G==0: state (16, 4096):      D = G x S        strideK = 4096
//   G==1: state (16, 16, 256):   per-h GEMMs      strideK = 256
//   G==2: state (256, 16, 16):   per-h GEMMs      strideK = 16
//   G==3: state (4096, 16):      D = S x G^T      (A = state rows)
// gfrag holds the gate operand fragment (A for G<3, B==G^T for G==3); it is
// invariant across a wave's 8 tiles, so it is loaded once by the caller.
// `wave` is an SGPR (readfirstlane) so the tile loop is fully scalar control
// flow and EXEC is never masked around the WMMA ops.
// ---------------------------------------------------------------------------
template <int G>
__device__ __forceinline__ void apply_group(float* __restrict__ st,
                                            const v2f (&gfrag)[4],
                                            int wave, int lane)
{
  const int m16  = lane & 15;           // M (or N) index inside fragment
  const int koff = (lane >> 4) << 1;    // lanes 16-31 hold K = 2,3
  const int hi8  = (lane >> 4) << 3;    // D reg r -> M = r + hi8

#pragma unroll 2
  for (int t = wave; t < 256; t += 32) {
    v8f d = {0.f, 0.f, 0.f, 0.f, 0.f, 0.f, 0.f, 0.f};

    if constexpr (G == 3) {
      const int r0 = t << 4;            // scalar
      v2f a[4];
#pragma unroll
      for (int i = 0; i < 4; ++i) {
        const int k = (i << 2) + koff;
        a[i].x = st[(r0 + m16) * 16 + k];
        a[i].y = st[(r0 + m16) * 16 + k + 1];
      }
#pragma unroll
      for (int i = 0; i < 4; ++i)
        d = __builtin_amdgcn_wmma_f32_16x16x4_f32(false, a[i], false, gfrag[i],
                                                  (short)0, d, false, false);
#pragma unroll
      for (int r = 0; r < 8; ++r)
        st[(r0 + hi8 + r) * 16 + m16] = d[r];
    } else {
      constexpr int sK = (G == 0) ? 4096 : (G == 1) ? 256 : 16;
      const int base = (G == 0) ? (t << 4)                                // scalar
                     : (G == 1) ? (((t >> 4) << 12) + ((t & 15) << 4))
                     :            (t << 8);
      v2f b[4];
#pragma unroll
      for (int i = 0; i < 4; ++i) {
        const int k = (i << 2) + koff;
        b[i].x = st[base + k * sK + m16];
        b[i].y = st[base + (k + 1) * sK + m16];
      }
#pragma unroll
      for (int i = 0; i < 4; ++i)
        d = __builtin_amdgcn_wmma_f32_16x16x4_f32(false, gfrag[i], false, b[i],
                                                  (short)0, d, false, false);
#pragma unroll
      for (int r = 0; r < 8; ++r)
        st[base + (hi8 + r) * sK + m16] = d[r];
    }
  }
}

// Build the fused gate for group G from per-qubit (cos, sin) tables, load the
// per-lane gate fragment, and apply to the state. Barriers included.
template <int G>
__device__ __forceinline__ void do_group(float* __restrict__ st,
                                         float* __restrict__ gate,
                                         const float* __restrict__ cq,
                                         const float* __restrict__ sq,
                                         int tid, int wave, int lane)
{
  // G[j][k] = prod_t RY(theta_{4G+t})[j_{3-t}, k_{3-t}],  M=[[c,-s],[s,c]]
  if (tid < 256) {
    const int j = tid >> 4, k = tid & 15;
    float val = 1.0f;
#pragma unroll
    for (int t = 0; t < 4; ++t) {
      const int q  = (G << 2) + t;
      const int jb = (j >> (3 - t)) & 1;
      const int kb = (k >> (3 - t)) & 1;
      val *= (jb == kb) ? cq[q] : ((jb == 0) ? -sq[q] : sq[q]);
    }
    gate[j * 17 + k] = val;
  }
  __syncthreads();

  // Gate fragment: gate[m16*17 + k] serves as A[m][k] (G<3) and as
  // B[k][n] = G^T (G==3) -- identical per-lane expression.
  const int m16  = lane & 15;
  const int koff = (lane >> 4) << 1;
  v2f gfrag[4];
#pragma unroll
  for (int i = 0; i < 4; ++i) {
    const int k = (i << 2) + koff;
    gfrag[i].x = gate[m16 * 17 + k];
    gfrag[i].y = gate[m16 * 17 + k + 1];
  }

  apply_group<G>(st, gfrag, wave, lane);
  __syncthreads();
}

__global__ void __launch_bounds__(THREADS)
qsim_kernel(const float* __restrict__ x, const float* __restrict__ w,
            float* __restrict__ out)
{
  __shared__ float st[NSTATE];        // 256 KB state, LDS-resident
  __shared__ float gate[16 * 17];     // fused 16x16 gate, padded stride 17
  __shared__ float cq[NQ], sq[NQ];    // per-qubit cos/sin for current round

  const int tid  = threadIdx.x;
  const int bidx = blockIdx.x;
  // wave id is uniform across the wave: pin it to an SGPR so tile loops and
  // tile base addresses are scalar (no EXEC masking around WMMA).
  const int wave = __builtin_amdgcn_readfirstlane(tid >> 5);
  const int lane = tid & 31;

  // |0...0>
  for (int i = tid; i < NSTATE; i += THREADS) st[i] = 0.0f;
  if (tid == 0) st[0] = 1.0f;
  __syncthreads();

  // round 0: input RY layer (angles = x[b,:]); rounds 1..4: weight layers,
  // each followed by the CNOT ladder == Gray-code gather m -> m ^ (m>>1).
  for (int round = 0; round < 5; ++round) {
    const float* ang = (round == 0) ? (x + bidx * NQ) : (w + (round - 1) * NQ);
    if (tid < NQ) {
      const float h = 0.5f * ang[tid];
      cq[tid] = __cosf(h);
      sq[tid] = __sinf(h);
    }
    __syncthreads();

    do_group<0>(st, gate, cq, sq, tid, wave, lane);
    do_group<1>(st, gate, cq, sq, tid, wave, lane);
    do_group<2>(st, gate, cq, sq, tid, wave, lane);
    do_group<3>(st, gate, cq, sq, tid, wave, lane);

    if (round > 0) {
      // CNOT chain (ctrl=i, tgt=i+1, i=0..14) composes to the Gray-code
      // permutation: out[m] = in[m ^ (m >> 1)]. Register-stage 64/thread.
      float tmp[64];
#pragma unroll
      for (int i = 0; i < 64; ++i) {
        const int m = tid + (i << 10);
        tmp[i] = st[m ^ (m >> 1)];
      }
      __syncthreads();
#pragma unroll
      for (int i = 0; i < 64; ++i)
        st[tid + (i << 10)] = tmp[i];
      __syncthreads();
    }
  }

  // reference returns state[:, :16]
  if (tid < NQ) out[bidx * NQ + tid] = st[tid];
}

extern "C" void kernel_launch(void* const* d_in, const int* in_sizes, int n_in,
                              void* d_out, int out_size, void* d_ws, size_t ws_size,
                              hipStream_t stream)
{
  const float* x = (const float*)d_in[0];   // (512, 16) f32
  const float* w = (const float*)d_in[1];   // (4, 16)  f32
  float* out = (float*)d_out;               // (512, 16) f32

  const int B = in_sizes[0] / NQ;
  qsim_kernel<<<B, THREADS, 0, stream>>>(x, w, out);
}